// DRR_8641474200473
// MI455X (gfx1250) — compile-verified
//
#include <hip/hip_runtime.h>
#include <math.h>

// DRR via incremental Siddon ray marching on MI455X (gfx1250).
//
// No WMMA: scattered-gather ray integral, not a matrix contraction.
// Strategy for MI455X:
//  * 64MB volume fits entirely in 192MB L2 -> warm it with global_prefetch_b8.
//  * Replace the reference's per-ray 771-float sort with an O(1)-state 3-way
//    merge of the per-axis alpha progressions (identical merged order).
//  * 40k rays = only 1250 waves -> too few to hide L2 gather latency on a
//    ~1000-SIMD chip. Split each ray across Q=8 lanes by exact alpha-bucket
//    partition (bit-identical thresholds per ray), march ~96 steps per lane,
//    reduce with ds_permute (__shfl_xor) inside each aligned 8-lane group.

#define IMG_H 200
#define IMG_W 200
#define DELX 1.5f
#define DELY 1.5f
#define NXV 256
#define NYV 256
#define NZV 256
#define RAY_EPS 1e-8f
#define F_INF __builtin_inff()
#define QSPLIT 8          // lanes per ray (power of 2, divides 32)

// ---------------------------------------------------------------------------
// Kernel 1: warm global L2 with the whole volume (64 MB fits in 192 MB).
// One 64B cacheline per thread -> global_prefetch_b8.
// ---------------------------------------------------------------------------
__global__ __launch_bounds__(256) void drr_prefetch_volume(
    const float* __restrict__ vol, unsigned int nlines) {
  unsigned int line = blockIdx.x * blockDim.x + threadIdx.x;
  if (line < nlines) {
    __builtin_prefetch(vol + (size_t)line * 16u, 0, 3);  // 16 floats = 64B
  }
}

// ---------------------------------------------------------------------------
// Kernel 2: QSPLIT lanes per ray, incremental Siddon march per alpha bucket.
// ---------------------------------------------------------------------------
__global__ __launch_bounds__(256) void drr_siddon_kernel(
    const float* __restrict__ volume,       // 256^3, row-major [x][y][z]
    const float* __restrict__ spacing,      // 3
    const float* __restrict__ sdr,          // 1
    const float* __restrict__ rotations,    // 3
    const float* __restrict__ translations, // 3
    float* __restrict__ out)                // H*W
{
  const int gid = blockIdx.x * blockDim.x + threadIdx.x;
  const int n = gid / QSPLIT;               // ray id
  const int q = gid & (QSPLIT - 1);         // bucket id within ray
  if (n >= IMG_H * IMG_W) return;
  const int pi = n / IMG_W;                 // row (t index)
  const int pj = n % IMG_W;                 // col (s index)

  // ---- rotation matrix R = Rz(theta) @ Ry(phi) @ Rx(gamma) (uniform) ----
  float st, ct, sp, cp, sg, cg;
  sincosf(rotations[0], &st, &ct);
  sincosf(rotations[1], &sp, &cp);
  sincosf(rotations[2], &sg, &cg);

  const float r0x = ct * cp, r0y = st * cp, r0z = -sp;      // R[:,0]
  const float b0x = -st * cg + ct * sp * sg;                // R[:,1]
  const float b0y =  ct * cg + st * sp * sg;
  const float b0z =  cp * sg;
  const float b1x =  st * sg + ct * sp * cg;                // R[:,2]
  const float b1y = -ct * sg + st * sp * cg;
  const float b1z =  cp * cg;

  const float S  = sdr[0];
  const float tx = translations[0], ty = translations[1], tz = translations[2];

  const float srcx = S * r0x + tx;
  const float srcy = S * r0y + ty;
  const float srcz = S * r0z + tz;
  const float cxx = -S * r0x + tx, cyy = -S * r0y + ty, czz = -S * r0z + tz;

  const float tc = (float)(pi - (IMG_H / 2 - 1)) * DELX;
  const float sc = (float)(pj - (IMG_W / 2 - 1)) * DELY;

  const float tgx = tc * b0x + sc * b1x + cxx;
  const float tgy = tc * b0y + sc * b1y + cyy;
  const float tgz = tc * b0z + sc * b1z + czz;

  const float sddx = tgx - srcx + RAY_EPS;
  const float sddy = tgy - srcy + RAY_EPS;
  const float sddz = tgz - srcz + RAY_EPS;

  const float spx = spacing[0], spy = spacing[1], spz = spacing[2];

  const float invx = 1.0f / sddx;
  const float invy = 1.0f / sddy;
  const float invz = 1.0f / sddz;

  const float a0x = (0.0f - srcx) * invx, a1x = ((float)NXV * spx - srcx) * invx;
  const float a0y = (0.0f - srcy) * invy, a1y = ((float)NYV * spy - srcy) * invy;
  const float a0z = (0.0f - srcz) * invz, a1z = ((float)NZV * spz - srcz) * invz;

  const float amin = fmaxf(fmaxf(fminf(a0x, a1x), fminf(a0y, a1y)), fminf(a0z, a1z));
  const float amax = fminf(fminf(fmaxf(a0x, a1x), fmaxf(a0y, a1y)), fmaxf(a0z, a1z));

  // ---- alpha-bucket thresholds (bit-identical across the 8 lanes of a ray) --
  const float range = amax - amin;
  const float tq   = (q == 0) ? amin
                   : fmaf(range, (float)q * (1.0f / QSPLIT), amin);
  const float tend = (q == QSPLIT - 1) ? amax
                   : fmaf(range, (float)(q + 1) * (1.0f / QSPLIT), amin);

  // ---- per-axis init: first plane k with alpha(k) > tq (strict), plus the
  //      predecessor alpha (largest plane alpha <= tq) for the left endpoint.
  // alpha(k) = (k*sp - src) * inv ; monotonic in k (dir = sign(inv))
#define AXIS_INIT(K, D, NEXT, PREV, INV, SDD, SRC, SP, NPL)                    \
  int K; int D; float NEXT; float PREV;                                        \
  do {                                                                         \
    if (!(fabsf(INV) <= 3.0e38f)) { K = 0; D = 0; NEXT = F_INF; PREV = -F_INF; break; } \
    float kf_ = fmaf(tq, SDD, SRC) / (SP);                                     \
    kf_ = fminf(fmaxf(kf_, -4.0f), (float)(NPL) + 4.0f);                       \
    int k_;                                                                    \
    if (INV > 0.0f) {                                                          \
      D = 1;                                                                   \
      k_ = (int)floorf(kf_) + 1;                                               \
      if (k_ < 0) k_ = 0;                                                      \
      if (k_ > (NPL) + 1) k_ = (NPL) + 1;                                      \
      while (k_ <= (NPL) && ((float)k_ * (SP) - (SRC)) * (INV) <= tq) ++k_;    \
      while (k_ > 0 && ((float)(k_ - 1) * (SP) - (SRC)) * (INV) > tq) --k_;    \
    } else {                                                                   \
      D = -1;                                                                  \
      k_ = (int)ceilf(kf_) - 1;                                                \
      if (k_ > (NPL)) k_ = (NPL);                                              \
      if (k_ < -1) k_ = -1;                                                    \
      while (k_ >= 0 && ((float)k_ * (SP) - (SRC)) * (INV) <= tq) --k_;        \
      while (k_ < (NPL) && ((float)(k_ + 1) * (SP) - (SRC)) * (INV) > tq) ++k_;\
    }                                                                          \
    K = k_;                                                                    \
    NEXT = ((unsigned)K <= (unsigned)(NPL))                                    \
               ? ((float)K * (SP) - (SRC)) * (INV) : F_INF;                    \
    const int kp_ = K - D;                                                     \
    PREV = ((unsigned)kp_ <= (unsigned)(NPL))                                  \
               ? ((float)kp_ * (SP) - (SRC)) * (INV) : -F_INF;                 \
  } while (0)

  AXIS_INIT(kx, dx, ax, px, invx, sddx, srcx, spx, NXV);
  AXIS_INIT(ky, dy, ay, py, invy, sddy, srcy, spy, NYV);
  AXIS_INIT(kz, dz, az, pz, invz, sddz, srcz, spz, NZV);
#undef AXIS_INIT

  // left endpoint: merged predecessor of this bucket (clamped to amin)
  float aprev = fmaxf(amin, fmaxf(fmaxf(px, py), pz));

  const float rspx = 1.0f / spx, rspy = 1.0f / spy, rspz = 1.0f / spz;

  // ---- 3-way merge march over this lane's bucket: pop min(next) <= tend ----
  float acc = 0.0f;
  int guard = NXV + NYV + NZV + 8;

  while (guard-- > 0) {
    const float an = fminf(fminf(ax, ay), az);
    if (!(an <= tend)) break;

    const float stp = an - aprev;
    const float mid = 0.5f * (aprev + an);

    const float x = fmaf(mid, sddx, srcx) * rspx;
    const float y = fmaf(mid, sddy, srcy) * rspy;
    const float z = fmaf(mid, sddz, srcz) * rspz;
    int ix = (int)x; ix = min(max(ix, 0), NXV - 1);
    int iy = (int)y; iy = min(max(iy, 0), NYV - 1);
    int iz = (int)z; iz = min(max(iz, 0), NZV - 1);

    // vol = flip(volume, axis 0): addr = ((255-ix)*256 + iy)*256 + iz
    const int addr = ((((NXV - 1 - ix) << 8) + iy) << 8) + iz;
    const float v = volume[addr];
    acc = fmaf(v, stp, acc);

    // branchless advance of the popped axis (divergent across lanes)
    const bool mx = (an == ax);
    const bool my = !mx && (an == ay);
    const bool mz = !mx && !my;
    kx = mx ? kx + dx : kx;
    ky = my ? ky + dy : ky;
    kz = mz ? kz + dz : kz;
    ax = mx ? (((unsigned)kx <= (unsigned)NXV)
                   ? ((float)kx * spx - srcx) * invx : F_INF) : ax;
    ay = my ? (((unsigned)ky <= (unsigned)NYV)
                   ? ((float)ky * spy - srcy) * invy : F_INF) : ay;
    az = mz ? (((unsigned)kz <= (unsigned)NZV)
                   ? ((float)kz * spz - srcz) * invz : F_INF) : az;
    aprev = an;
  }

  // ---- reduce the 8 bucket partial sums (aligned 8-lane group, wave32) ----
  acc += __shfl_xor(acc, 1, 32);
  acc += __shfl_xor(acc, 2, 32);
  acc += __shfl_xor(acc, 4, 32);

  if (q == 0) {
    const float nrm = sqrtf(sddx * sddx + sddy * sddy + sddz * sddz);
    out[n] = acc * nrm;
  }
}

// ---------------------------------------------------------------------------
extern "C" void kernel_launch(void* const* d_in, const int* in_sizes, int n_in,
                              void* d_out, int out_size, void* d_ws, size_t ws_size,
                              hipStream_t stream) {
  (void)in_sizes; (void)n_in; (void)d_ws; (void)ws_size; (void)out_size;

  const float* volume       = (const float*)d_in[0];
  const float* spacing      = (const float*)d_in[1];
  const float* sdr          = (const float*)d_in[2];
  const float* rotations    = (const float*)d_in[3];
  const float* translations = (const float*)d_in[4];
  float* out                = (float*)d_out;

  // Warm L2 with the full volume (256^3 floats / 16 per 64B line).
  const unsigned int nlines = (NXV * NYV * NZV) / 16u;
  drr_prefetch_volume<<<(nlines + 255u) / 256u, 256, 0, stream>>>(volume, nlines);

  const int nthreads = IMG_H * IMG_W * QSPLIT;
  drr_siddon_kernel<<<(nthreads + 255) / 256, 256, 0, stream>>>(
      volume, spacing, sdr, rotations, translations, out);
}